// CombinedRotaryEmbedding_43533788512529
// MI455X (gfx1250) — compile-verified
//
#include <hip/hip_runtime.h>
#include <hip/hip_bf16.h>

typedef __attribute__((ext_vector_type(2))) float v2f;
typedef __attribute__((ext_vector_type(4))) float v4f;
typedef __attribute__((ext_vector_type(8))) float v8f;

// ---------------------------------------------------------------------------
// Setup kernel: collapse the 32-step Givens scan into a 64x64 matrix G,
// multiply by r_matrix, and store the combined 64x64 operator into workspace
// in "row-pair interleaved" layout:
//     A_pk[(r>>1)*128 + c*2 + (r&1)] = A[r][c]
// so a WMMA B-fragment (two consecutive K rows, same column) is one LDS
// 64-bit read, and an even/odd column *pair* of fragments is one 128-bit read.
// One block, 64 threads; thread t owns row t of G (rows are independent).
// ---------------------------------------------------------------------------
__global__ void build_combined_kernel(const float* __restrict__ thetas,
                                      const float* __restrict__ theta_scale,
                                      const float* __restrict__ r_matrix,
                                      const int*   __restrict__ r_pairs,
                                      float* __restrict__ A_pk) {
    __shared__ float G[64 * 64];
    const int t = threadIdx.x;          // row of G, 0..63
    for (int c = 0; c < 64; ++c) G[t * 64 + c] = (c == t) ? 1.0f : 0.0f;

    const float scale = theta_scale[0];
    for (int k = 0; k < 32; ++k) {      // sequential rotations, rows independent
        const float th  = thetas[k] * scale;
        const float cth = cosf(th);
        const float sth = sinf(th);
        const int i = r_pairs[2 * k + 0];
        const int j = r_pairs[2 * k + 1];
        const float gi = G[t * 64 + i];
        const float gj = G[t * 64 + j];
        G[t * 64 + i] =  gi * cth + gj * sth;
        G[t * 64 + j] = -gi * sth + gj * cth;
    }
    // row t of (G @ r_matrix), stored interleaved
    for (int c = 0; c < 64; ++c) {
        float acc = 0.0f;
        for (int k = 0; k < 64; ++k)
            acc = fmaf(G[t * 64 + k], r_matrix[k * 64 + c], acc);
        A_pk[(t >> 1) * 128 + c * 2 + (t & 1)] = acc;
    }
}

// ---------------------------------------------------------------------------
// Main kernel. Per 16-row tile (rows share one sequence position s):
//   y[16x64] = X[16x64] @ A[64x64]  via 16 k-steps x 4 n-tiles of
//   V_WMMA_F32_16X16X4_F32.
// Column permutation trick: n-tiles 0/1 compute y's EVEN columns
// (y[2d], d = 16h + lane), n-tiles 2/3 the matching ODD columns (y[2d+1]),
// lane-aligned.  The RoPE pair therefore lives in one lane (no shuffles) and
// output columns are lane-contiguous (fully coalesced non-temporal stores):
//   out[.., d]    = y[2d]*cos - y[2d+1]*sin      (d = 16h + lane)
//   out[.., 32+d] = y[2d]*sin + y[2d+1]*cos
// ---------------------------------------------------------------------------
__global__ __launch_bounds__(256) void rope_gemm_wmma(
    const float* __restrict__ x,
    const float* __restrict__ inv_freq,
    const float* __restrict__ A_pk,
    float* __restrict__ out,
    int ntiles) {
    __shared__ __align__(16) float Alds[4096];   // 16 KB combined matrix
    {
        const float4* src = (const float4*)A_pk;
        float4*       dst = (float4*)Alds;
#pragma unroll
        for (int i = 0; i < 4; ++i)
            dst[threadIdx.x + 256 * i] = src[threadIdx.x + 256 * i];
    }
    __syncthreads();

    const int lane = threadIdx.x & 31;
    const int half = lane >> 4;     // K-half / M-half selector
    const int mcol = lane & 15;     // A-operand row M; B/C/D column N
    const int nwaves = gridDim.x * (blockDim.x >> 5);
    const int wave0  = blockIdx.x * (blockDim.x >> 5) + (threadIdx.x >> 5);

    // per-lane RoPE frequencies: freq index d = 16h + mcol, h = 0,1
    const float invf0 = inv_freq[mcol];
    const float invf1 = inv_freq[16 + mcol];

    for (int tile = wave0; tile < ntiles; tile += nwaves) {   // wave-uniform loop
        const float* xrow = x + (size_t)tile * 1024 + (size_t)mcol * 64 + 2 * half;
        v8f acc[4] = {};
#pragma unroll
        for (int k = 0; k < 16; ++k) {
            // A fragment: lane holds X[m, 4k + 2*half .. +1]  (contiguous pair)
            v2f a = __builtin_nontemporal_load((const v2f*)(xrow + 4 * k));
            const float* bb = Alds + (2 * k + half) * 128;
            // One b128 LDS read = fragments for y-cols {2d, 2d+1}, d = mcol:
            v4f blo = *(const v4f*)(bb + 4 * mcol);        // d in [0,16)
            v4f bhi = *(const v4f*)(bb + 64 + 4 * mcol);   // d in [16,32)
            v2f b0 = __builtin_shufflevector(blo, blo, 0, 1);  // even col 2*mcol
            v2f b2 = __builtin_shufflevector(blo, blo, 2, 3);  // odd  col 2*mcol+1
            v2f b1 = __builtin_shufflevector(bhi, bhi, 0, 1);  // even col 32+2*mcol
            v2f b3 = __builtin_shufflevector(bhi, bhi, 2, 3);  // odd  col 33+2*mcol
            acc[0] = __builtin_amdgcn_wmma_f32_16x16x4_f32(false, a, false, b0,
                                                           (short)0, acc[0], false, false);
            acc[1] = __builtin_amdgcn_wmma_f32_16x16x4_f32(false, a, false, b1,
                                                           (short)0, acc[1], false, false);
            acc[2] = __builtin_amdgcn_wmma_f32_16x16x4_f32(false, a, false, b2,
                                                           (short)0, acc[2], false, false);
            acc[3] = __builtin_amdgcn_wmma_f32_16x16x4_f32(false, a, false, b3,
                                                           (short)0, acc[3], false, false);
        }

        // RoPE epilogue: shuffle-free, lane-contiguous non-temporal stores
        const float pos = (float)(tile & 4095);   // s = tile % S
        float* obase = out + (size_t)tile * 1024 + (size_t)(8 * half) * 64 + mcol;
#pragma unroll
        for (int h = 0; h < 2; ++h) {
            const float ang = pos * (h ? invf1 : invf0);
            const float cs = __cosf(ang);
            const float sn = __sinf(ang);
#pragma unroll
            for (int v = 0; v < 8; ++v) {
                const float ye = acc[h][v];       // y[2d]
                const float yo = acc[h + 2][v];   // y[2d+1]
                __builtin_nontemporal_store(fmaf(ye, cs, -(yo * sn)),
                                            obase + (size_t)v * 64 + 16 * h);
                __builtin_nontemporal_store(fmaf(ye, sn, yo * cs),
                                            obase + (size_t)v * 64 + 32 + 16 * h);
            }
        }
    }
}

extern "C" void kernel_launch(void* const* d_in, const int* in_sizes, int n_in,
                              void* d_out, int out_size, void* d_ws, size_t ws_size,
                              hipStream_t stream) {
    const float* x           = (const float*)d_in[0];
    const float* thetas      = (const float*)d_in[1];
    const float* theta_scale = (const float*)d_in[2];
    const float* r_matrix    = (const float*)d_in[3];
    const float* inv_freq    = (const float*)d_in[4];
    const int*   r_pairs     = (const int*)d_in[5];
    float*       out         = (float*)d_out;
    float*       A_pk        = (float*)d_ws;   // 16 KB combined-operator scratch

    // 1) fold Givens scan + r_matrix into one 64x64 operator (tiny, one block)
    build_combined_kernel<<<1, 64, 0, stream>>>(thetas, theta_scale, r_matrix,
                                                r_pairs, A_pk);

    // 2) memory-bound GEMM+RoPE: 16-row tiles, wave-uniform grid-stride
    const int ntiles = in_sizes[0] / 1024;     // (B*S*N_HEAD*64) / (16*64)
    rope_gemm_wmma<<<1024, 256, 0, stream>>>(x, inv_freq, A_pk, out, ntiles);
}